// SemanticConsistencyLoss_41764261986808
// MI455X (gfx1250) — compile-verified
//
#include <hip/hip_runtime.h>
#include <float.h>

typedef __attribute__((ext_vector_type(2))) float v2f;
typedef __attribute__((ext_vector_type(8))) float v8f;

__global__ void SemLoss_init_acc(unsigned int* acc) {
    if (threadIdx.x == 0 && blockIdx.x == 0) *acc = 0u;
}

// One wave per 16-row tile of the query set. For each 16-column tile of the
// canonical set, compute D = A*B + C with V_WMMA_F32_16X16X4_F32 where
// C = -0.5*||y_j||^2 broadcast down each column, so D[i,j] = x_i.y_j - ||y_j||^2/2.
// argmax_j D == argmin_j ||x_i - y_j||^2.
// Column-tile loop is unrolled x4 with grouped loads so the global_load_b96s
// for 4 tiles are in flight together and overlap with WMMA/VALU of earlier
// tiles instead of a full s_wait_loadcnt 0 stall per tile.
__global__ __launch_bounds__(256) void SemLoss_nn_kernel(
    const float* __restrict__ x,      // [N,3] queries
    const float* __restrict__ y,      // [N,3] canonical
    const int*   __restrict__ labels, // [N]
    unsigned int* __restrict__ acc,
    int N)
{
    const int lane = threadIdx.x & 31;
    const int wave = threadIdx.x >> 5;
    const int numTiles = (N + 15) >> 4;
    const int rowTile = blockIdx.x * 8 + wave;
    if (rowTile >= numTiles) return;          // wave-uniform exit

    const int sub = lane & 15;                // position within 16-lane half
    const int hi  = lane >> 4;                // 0: lanes 0-15, 1: lanes 16-31

    // --- A tile: 16x4 f32, lane layout per ISA (VGPR0: K0|K2, VGPR1: K1|K3) ---
    int r  = rowTile * 16 + sub;
    int rc = (r < N) ? r : (N - 1);
    float xx = x[3 * rc + 0], xy = x[3 * rc + 1], xz = x[3 * rc + 2];
    v2f a;
    a.x = hi ? xz : xx;                       // K=0 (x) / K=2 (z)
    a.y = hi ? 0.0f : xy;                     // K=1 (y) / K=3 (pad 0)

    float best[8];
    int   bidx[8];
#pragma unroll
    for (int v = 0; v < 8; ++v) { best[v] = -FLT_MAX; bidx[v] = 0; }

    constexpr int U = 4;                      // column tiles in flight
    const int numTilesU = (numTiles / U) * U;

    int t = 0;
    for (; t < numTilesU; t += U) {
        float yx[U], yy[U], yz[U];
        bool  val[U];
        // Phase 1: issue all U loads (independent dests -> batched vmem)
#pragma unroll
        for (int u = 0; u < U; ++u) {
            int c  = (t + u) * 16 + sub;
            val[u] = (c < N);
            int ci = val[u] ? c : 0;
            yx[u] = y[3 * ci + 0];
            yy[u] = y[3 * ci + 1];
            yz[u] = y[3 * ci + 2];
        }
        // Phase 2: U WMMAs + argmax updates
#pragma unroll
        for (int u = 0; u < U; ++u) {
            v2f b;
            b.x = hi ? yz[u] : yx[u];
            b.y = hi ? 0.0f : yy[u];
            float cval = val[u]
                ? (-0.5f * (yx[u] * yx[u] + yy[u] * yy[u] + yz[u] * yz[u]))
                : -1.0e30f;
            v8f cm;
#pragma unroll
            for (int v = 0; v < 8; ++v) cm[v] = cval;

            v8f d = __builtin_amdgcn_wmma_f32_16x16x4_f32(
                false, a, false, b, (short)0, cm, false, false);

            int c = (t + u) * 16 + sub;
#pragma unroll
            for (int v = 0; v < 8; ++v) {
                if (d[v] > best[v]) { best[v] = d[v]; bidx[v] = c; }
            }
        }
    }
    // Remainder tiles (< U of them)
    for (; t < numTiles; ++t) {
        int  c     = t * 16 + sub;
        bool valid = (c < N);
        int  ci    = valid ? c : 0;
        float yx = y[3 * ci + 0], yy = y[3 * ci + 1], yz = y[3 * ci + 2];
        v2f b;
        b.x = hi ? yz : yx;
        b.y = hi ? 0.0f : yy;
        float cval = valid ? (-0.5f * (yx * yx + yy * yy + yz * yz)) : -1.0e30f;
        v8f cm;
#pragma unroll
        for (int v = 0; v < 8; ++v) cm[v] = cval;

        v8f d = __builtin_amdgcn_wmma_f32_16x16x4_f32(
            false, a, false, b, (short)0, cm, false, false);

#pragma unroll
        for (int v = 0; v < 8; ++v) {
            if (d[v] > best[v]) { best[v] = d[v]; bidx[v] = c; }
        }
    }

    // Reduce (max value, tie -> min index) across each 16-lane half.
    // VGPR v holds row (rowTile*16 + v) in lanes 0-15, row (+8) in lanes 16-31.
    unsigned int cnt = 0;
#pragma unroll
    for (int v = 0; v < 8; ++v) {
        float val = best[v];
        int   idx = bidx[v];
        for (int off = 8; off >= 1; off >>= 1) {
            float ov = __shfl_xor(val, off, 32);
            int   oi = __shfl_xor(idx, off, 32);
            if (ov > val || (ov == val && oi < idx)) { val = ov; idx = oi; }
        }
        if (sub == 0) {
            int row = rowTile * 16 + v + hi * 8;
            if (row < N) cnt += (labels[idx] != labels[row]) ? 1u : 0u;
        }
    }
    cnt += __shfl_xor(cnt, 16, 32);           // combine lane0 + lane16
    if (lane == 0) atomicAdd(acc, cnt);
}

__global__ void SemLoss_finalize(const unsigned int* acc, float* out, int N) {
    if (threadIdx.x == 0 && blockIdx.x == 0)
        out[0] = (float)(*acc) / (float)N;
}

extern "C" void kernel_launch(void* const* d_in, const int* in_sizes, int n_in,
                              void* d_out, int out_size, void* d_ws, size_t ws_size,
                              hipStream_t stream) {
    const float* x      = (const float*)d_in[0];   // mean_3d      [N,3]
    const float* y      = (const float*)d_in[1];   // mean_3d_cano [N,3]
    const int*   labels = (const int*)d_in[2];     // segm_labels  [N]
    float*       out    = (float*)d_out;
    unsigned int* acc   = (unsigned int*)d_ws;

    const int N = in_sizes[0] / 3;                 // 20000
    const int numTiles = (N + 15) >> 4;            // 1250 row tiles
    const int blocks = (numTiles + 7) / 8;         // 8 waves (256 thr) per block

    SemLoss_init_acc<<<1, 1, 0, stream>>>(acc);
    SemLoss_nn_kernel<<<blocks, 256, 0, stream>>>(x, y, labels, acc, N);
    SemLoss_finalize<<<1, 1, 0, stream>>>(acc, out, N);
}